// MMDLoss_20469814132952
// MI455X (gfx1250) — compile-verified
//
#include <hip/hip_runtime.h>
#include <hip/hip_bf16.h>

typedef __attribute__((ext_vector_type(2))) float v2f;
typedef __attribute__((ext_vector_type(8))) float v8f;
typedef __attribute__((ext_vector_type(4))) unsigned int v4u;
typedef __attribute__((ext_vector_type(8))) int v8i;
typedef __attribute__((ext_vector_type(4))) int v4i;

#define N_HALF 4096
#define D      512
#define M_TOT  8192
#define BM     128
#define BN     128
#define BK     16
#define LDK    20      // padded LDS stride (floats); TDM pad fields reproduce this
#define NTILE  64      // 8192 / 128
#define TILEF  (BM * LDK)   // floats per staged tile (2560)

#if __has_builtin(__builtin_amdgcn_tensor_load_to_lds) && __has_builtin(__builtin_amdgcn_s_wait_tensorcnt)
#define USE_TDM 1
#else
#define USE_TDM 0
#endif

// ---------------- ws layout (floats) ----------------
#define WS_SQ      0
#define WS_COLSUM  8192
#define WS_BW      8704
#define WS_PART    8960

__device__ __forceinline__ const float* row_base(const float* fs, const float* ft, int r) {
  return (r < N_HALF) ? fs + (size_t)r * D : ft + (size_t)(r - N_HALF) * D;
}

#if USE_TDM
// Issue one TDM descriptor: load a 128-row x 16-col fp32 tile (row stride 512)
// into LDS at lds_byte_off, with HW padding of 4 DWORDs after every 16 DWORDs
// (=> LDS row stride of 20 floats, matching LDK).
__device__ __forceinline__ void tdm_load_tile(const float* gsrc, unsigned int lds_byte_off) {
  unsigned long long ga = (unsigned long long)(uintptr_t)gsrc;
  v4u g0;
  g0.x = 1u;                                        // count=1 (valid), user mode
  g0.y = lds_byte_off;                              // lds_addr
  g0.z = (unsigned int)(ga & 0xFFFFFFFFu);          // global_addr[31:0]
  g0.w = (unsigned int)((ga >> 32) & 0x01FFFFFFu)   // global_addr[56:32]
         | (2u << 30);                              // type = 2 ("image")
  v8i g1;
  g1[0] = (int)((2u << 16)        // data_size = 4 bytes
              | (1u << 20)        // pad_enable
              | (3u << 22)        // pad_interval: 16 DWORDs
              | (3u << 25));      // pad_amount:   4 DWORDs
  g1[1] = (int)(((unsigned)D & 0xFFFFu) << 16);     // tensor_dim0[15:0] = 512
  g1[2] = (int)(((unsigned)BM & 0xFFFFu) << 16);    // dim0[31:16]=0 | tensor_dim1 = 128
  g1[3] = (int)((unsigned)BK << 16);                // dim1[31:16]=0 | tile_dim0 = 16
  g1[4] = (int)BM;                                  // tile_dim1 = 128 | tile_dim2 = 0
  g1[5] = (int)D;                                   // tensor_dim0_stride[31:0] = 512
  g1[6] = 0;                                        // stride hi | tensor_dim1_stride lo
  g1[7] = 0;
  v4i z4 = {0, 0, 0, 0};                            // groups 2/3 unused (2-D tile)
  v8i z8 = {0, 0, 0, 0, 0, 0, 0, 0};
  __builtin_amdgcn_tensor_load_to_lds(g0, g1, z4, z4, z8, 0);
}
#endif

// ---------- Kernel 1: per-row squared norms (one wave per row) ----------
__global__ __launch_bounds__(256) void mmd_rowsq(const float* __restrict__ fs,
                                                 const float* __restrict__ ft,
                                                 float* __restrict__ sq) {
  int wave = threadIdx.x >> 5;
  int lane = threadIdx.x & 31;
  int r = blockIdx.x * 8 + wave;
  const float4* b4 = (const float4*)row_base(fs, ft, r);
  float s = 0.0f;
#pragma unroll
  for (int t = 0; t < 4; ++t) {
    float4 v = b4[lane + 32 * t];
    s += v.x * v.x + v.y * v.y + v.z * v.z + v.w * v.w;
  }
#pragma unroll
  for (int o = 16; o > 0; o >>= 1) s += __shfl_down(s, o, 32);
  if (lane == 0) sq[r] = s;
}

// ---------- Kernel 2: column sums (deterministic) ----------
__global__ __launch_bounds__(256) void mmd_colsum(const float* __restrict__ fs,
                                                  const float* __restrict__ ft,
                                                  float* __restrict__ cs) {
  int c = blockIdx.x * 256 + threadIdx.x;
  float s = 0.0f;
#pragma unroll 8
  for (int r = 0; r < N_HALF; ++r) s += fs[(size_t)r * D + c];
#pragma unroll 8
  for (int r = 0; r < N_HALF; ++r) s += ft[(size_t)r * D + c];
  cs[c] = s;
}

// ---------- Kernel 3: bandwidth via closed form ----------
__global__ __launch_bounds__(256) void mmd_bw(const float* __restrict__ sq,
                                              const float* __restrict__ cs,
                                              float* __restrict__ bwout) {
  __shared__ float r1[256], r2[256];
  int tid = threadIdx.x;
  float a = 0.0f, b = 0.0f;
  for (int i = tid; i < M_TOT; i += 256) a += sq[i];
  for (int i = tid; i < D; i += 256) { float v = cs[i]; b += v * v; }
  r1[tid] = a; r2[tid] = b;
  __syncthreads();
  for (int k = 128; k > 0; k >>= 1) {
    if (tid < k) { r1[tid] += r1[tid + k]; r2[tid] += r2[tid + k]; }
    __syncthreads();
  }
  if (tid == 0) {
    float sumL2 = 2.0f * (float)M_TOT * r1[0] - 2.0f * r2[0];
    float bw = sumL2 / ((float)M_TOT * (float)M_TOT - (float)M_TOT);
    bwout[0] = bw * 0.25f;
  }
}

// ---------- Kernel 4: WMMA Gram tiles, TDM-double-buffered staging ----------
__global__ __launch_bounds__(256) void mmd_tiles(const float* __restrict__ fs,
                                                 const float* __restrict__ ft,
                                                 const float* __restrict__ sq,
                                                 const float* __restrict__ bwp,
                                                 float* __restrict__ partials) {
  const int ti = blockIdx.x;
  const int tj = blockIdx.y;
  const int pidx = tj * NTILE + ti;
  const int tid = threadIdx.x;
  if (ti > tj) {
    if (tid == 0) partials[pidx] = 0.0f;
    return;
  }

#if USE_TDM
  __shared__ float smem[2 * 2 * TILEF];   // [buf][A|B][BM*LDK]
#else
  __shared__ float smem[2 * TILEF];       // [A|B][BM*LDK]
#endif
  __shared__ float sqI[BM];
  __shared__ float sqJ[BN];
  __shared__ float red[256];

  const int i0 = ti * BM, j0 = tj * BN;
  const float* Abase = row_base(fs, ft, i0);
  const float* Bbase = row_base(fs, ft, j0);

  if (tid < 128) sqI[tid] = sq[i0 + tid];
  else           sqJ[tid - 128] = sq[j0 + (tid - 128)];

  const float bw = bwp[0];
  const float ninvbw = -1.0f / bw;

  const int lane = tid & 31;
  const int wave = tid >> 5;
  const int wm = wave & 1;
  const int wn = wave >> 1;
  const int h = lane >> 4;
  const int lr = lane & 15;

  v8f acc[4][2] = {};
  const int S = D / BK;   // 32 stages

#if USE_TDM
  const unsigned int smem_base = (unsigned int)(uintptr_t)&smem[0];
  const bool issuer = (tid < 32);          // wave 0 drives the TDM
  if (issuer) {
    tdm_load_tile(Abase, smem_base);
    tdm_load_tile(Bbase, smem_base + TILEF * 4u);
  }
#endif

  for (int s = 0; s < S; ++s) {
#if USE_TDM
    const int buf = s & 1;
    const float* sA = &smem[buf * 2 * TILEF];
    const float* sB = sA + TILEF;
    if (issuer) {
      if (s + 1 < S) {
        unsigned int nb = smem_base + (unsigned int)((buf ^ 1) * 2 * TILEF) * 4u;
        tdm_load_tile(Abase + (s + 1) * BK, nb);
        tdm_load_tile(Bbase + (s + 1) * BK, nb + TILEF * 4u);
        __builtin_amdgcn_s_wait_tensorcnt(2);   // in-order: stage s tiles landed
      } else {
        __builtin_amdgcn_s_wait_tensorcnt(0);
      }
    }
    __syncthreads();                            // publish buf to all waves
#else
    const float* sA = &smem[0];
    const float* sB = &smem[TILEF];
    const int k0 = s * BK;
#pragma unroll
    for (int rep = 0; rep < 2; ++rep) {
      int idx = tid + rep * 256;
      int row = idx >> 2;
      int kq = (idx & 3) << 2;
      float4 av = *(const float4*)(Abase + (size_t)row * D + k0 + kq);
      float4 bv = *(const float4*)(Bbase + (size_t)row * D + k0 + kq);
      *(float4*)&smem[row * LDK + kq] = av;
      *(float4*)&smem[TILEF + row * LDK + kq] = bv;
    }
    __syncthreads();
#endif

#pragma unroll
    for (int ks = 0; ks < BK / 4; ++ks) {
      // fp32 A/B fragment (16x4): lane l holds M=l&15, K = 2*(l>>4)+v
      const int kk = ks * 4 + 2 * h;
      v2f afrag[4], bfrag[2];
      const int rb = wm * 64 + lr;
#pragma unroll
      for (int a = 0; a < 4; ++a)
        afrag[a] = *(const v2f*)&sA[(rb + a * 16) * LDK + kk];
      const int cb = wn * 32 + lr;
#pragma unroll
      for (int b = 0; b < 2; ++b)
        bfrag[b] = *(const v2f*)&sB[(cb + b * 16) * LDK + kk];
#pragma unroll
      for (int a = 0; a < 4; ++a)
#pragma unroll
        for (int b = 0; b < 2; ++b)
          acc[a][b] = __builtin_amdgcn_wmma_f32_16x16x4_f32(
              false, afrag[a], false, bfrag[b], (short)0, acc[a][b], false, false);
    }
    __syncthreads();   // readers drained before this buffer is DMA-overwritten
  }

  // fused epilogue: L2 -> 5 Gaussian kernels -> signed partial sum
  const float sgn = ((i0 < N_HALF) == (j0 < N_HALF)) ? 1.0f : -1.0f;
  const float wt = (ti == tj) ? 1.0f : 2.0f;
  float local = 0.0f;
#pragma unroll
  for (int a = 0; a < 4; ++a) {
    const int rloc0 = wm * 64 + a * 16 + 8 * h;
#pragma unroll
    for (int b = 0; b < 2; ++b) {
      const float sqj = sqJ[wn * 32 + b * 16 + lr];
#pragma unroll
      for (int e = 0; e < 8; ++e) {
        float g = acc[a][b][e];
        float l2 = sqI[rloc0 + e] + sqj - 2.0f * g;
        float t = l2 * ninvbw;
        local += __expf(t) + __expf(t * 0.5f) + __expf(t * 0.25f) +
                 __expf(t * 0.125f) + __expf(t * 0.0625f);
      }
    }
  }
  red[tid] = local;
  __syncthreads();
  for (int k = 128; k > 0; k >>= 1) {
    if (tid < k) red[tid] += red[tid + k];
    __syncthreads();
  }
  if (tid == 0) partials[pidx] = red[0] * sgn * wt;
}

// ---------- Kernel 5: deterministic final reduction ----------
__global__ __launch_bounds__(256) void mmd_finalize(const float* __restrict__ partials,
                                                    float* __restrict__ out) {
  __shared__ float red[256];
  int tid = threadIdx.x;
  float s = 0.0f;
  for (int i = tid; i < NTILE * NTILE; i += 256) s += partials[i];
  red[tid] = s;
  __syncthreads();
  for (int k = 128; k > 0; k >>= 1) {
    if (tid < k) red[tid] += red[tid + k];
    __syncthreads();
  }
  if (tid == 0) out[0] = red[0] * (1.0f / ((float)N_HALF * (float)N_HALF));
}

extern "C" void kernel_launch(void* const* d_in, const int* in_sizes, int n_in,
                              void* d_out, int out_size, void* d_ws, size_t ws_size,
                              hipStream_t stream) {
  const float* fs = (const float*)d_in[0];
  const float* ft = (const float*)d_in[1];
  float* ws = (float*)d_ws;
  float* sq = ws + WS_SQ;
  float* cs = ws + WS_COLSUM;
  float* bw = ws + WS_BW;
  float* pp = ws + WS_PART;
  float* out = (float*)d_out;

  mmd_rowsq<<<M_TOT / 8, 256, 0, stream>>>(fs, ft, sq);
  mmd_colsum<<<2, 256, 0, stream>>>(fs, ft, cs);
  mmd_bw<<<1, 256, 0, stream>>>(sq, cs, bw);
  dim3 grid(NTILE, NTILE);
  mmd_tiles<<<grid, 256, 0, stream>>>(fs, ft, sq, bw, pp);
  mmd_finalize<<<1, 256, 0, stream>>>(pp, out);
}